// Sobel_24326694765137
// MI455X (gfx1250) — compile-verified
//
#include <hip/hip_runtime.h>
#include <math.h>

// ---------------- problem constants ----------------
#define Hd 4096
#define Wd 4096
#define PLANE (Hd * Wd)

#define TX 128            // output tile width
#define TY 32             // output tile height
#define GXh (TX + 8)      // 136  gray region (halo 4)
#define GYh (TY + 8)      // 40
#define BXh (TX + 6)      // 134  blur region (halo 3)
#define BYh (TY + 6)      // 38
#define MXh (TX + 4)      // 132  magnitude region (halo 2)
#define MYh (TY + 4)      // 36

#define NG (GXh * GYh)    // 5440
#define NB (BXh * BYh)    // 5092
#define NM (MXh * MYh)    // 4752
#define NHM (MYh * TX)    // 4608
#define NTHREADS 256

// Gaussian sigma=1: e=exp(-0.5); g = [e,1,e]/(1+2e)
#define GW0 0.27406862f
#define GW1 0.45186276f

// ---------------- native vector types ----------------
typedef int   v4i __attribute__((ext_vector_type(4)));
typedef float v4f __attribute__((ext_vector_type(4)));

typedef __attribute__((address_space(1))) int  as1_int;
typedef __attribute__((address_space(3))) int  as3_int;
typedef __attribute__((address_space(1))) v4i  as1_v4i;
typedef __attribute__((address_space(3))) v4i  as3_v4i;
typedef __attribute__((address_space(3))) void as3_void;

// ---------------- CDNA5 async global->LDS helpers ----------------
#if defined(__AMDGCN__) && __has_builtin(__builtin_amdgcn_global_load_async_to_lds_b32)
#define HAVE_ASYNC_B32 1
#else
#define HAVE_ASYNC_B32 0
#endif
#if defined(__AMDGCN__) && __has_builtin(__builtin_amdgcn_global_load_async_to_lds_b128)
#define HAVE_ASYNC_B128 1
#else
#define HAVE_ASYNC_B128 0
#endif

__device__ __forceinline__ unsigned lds_off32(const float* l) {
  return (unsigned)(unsigned long long)(as3_void*)l;
}

__device__ __forceinline__ void async_copy_b32(const float* g, float* l) {
#if HAVE_ASYNC_B32
  __builtin_amdgcn_global_load_async_to_lds_b32((as1_int*)g, (as3_int*)l, 0, 0);
#else
  asm volatile("global_load_async_to_lds_b32 %0, %1, off"
               :: "v"(lds_off32(l)), "v"(g) : "memory");
#endif
}

__device__ __forceinline__ void async_copy_b128(const float* g, float* l) {
#if HAVE_ASYNC_B128
  __builtin_amdgcn_global_load_async_to_lds_b128((as1_v4i*)g, (as3_v4i*)l, 0, 0);
#else
  asm volatile("global_load_async_to_lds_b128 %0, %1, off"
               :: "v"(lds_off32(l)), "v"(g) : "memory");
#endif
}

__device__ __forceinline__ void wait_async_zero() {
#if defined(__AMDGCN__) && __has_builtin(__builtin_amdgcn_s_wait_asynccnt)
  __builtin_amdgcn_s_wait_asynccnt(0);
#else
  asm volatile("s_wait_asynccnt 0" ::: "memory");
#endif
}

__device__ __forceinline__ void nt_store4(float* p, v4f v) {
#if defined(__AMDGCN__) && __has_builtin(__builtin_nontemporal_store)
  __builtin_nontemporal_store(v, (v4f*)p);
#else
  *(v4f*)p = v;
#endif
}

// reflect (no edge repeat): -1 -> 1, H -> H-2 ; valid for t in [-(H-1), 2H-2]
__device__ __forceinline__ int refl(int t) {
  t = (t < 0) ? -t : t;
  t = (t >= Hd) ? (2 * Hd - 2 - t) : t;
  return t;
}

__device__ __forceinline__ v4f max4(v4f a, v4f b) {
  v4f r;
  r.x = fmaxf(a.x, b.x); r.y = fmaxf(a.y, b.y);
  r.z = fmaxf(a.z, b.z); r.w = fmaxf(a.w, b.w);
  return r;
}

// ---------------- fused kernel ----------------
// LDS layout (all phases alias into a 3*NG float arena = 65,280 B):
//   phase 1 : S[0..NG)       = R stage   S[NG..2NG) = G stage   S[2NG..3NG) = B stage
//   phase 1b: gray written IN-PLACE over R stage (each thread rewrites its own elements)
//   phase 2 : blur  -> S[NG .. NG+NB)      (G stage dead)
//   phase 3 : mag   -> S[2NG .. 2NG+NM)    (B stage dead)
//   phase 4a: hmax  -> S[0 .. NHM)         (gray dead)
__global__ __launch_bounds__(NTHREADS)
void sobel_fused(const float* __restrict__ img, float* __restrict__ out) {
  __shared__ __align__(16) float S[3 * NG];

  const int tid  = threadIdx.x;
  const int col0 = blockIdx.x * TX;
  const int row0 = blockIdx.y * TY;

  const bool interior = (blockIdx.x > 0) && (blockIdx.x < gridDim.x - 1) &&
                        (blockIdx.y > 0) && (blockIdx.y < gridDim.y - 1);

  // ---- phase 1: stage RGB tile (+halo 4) into LDS via async loads ----
  if (interior) {
    // aligned b128 path: 3 planes * 40 rows * 34 float4 chunks = 4080
    const float* base = img + (long long)(row0 - 4) * Wd + (col0 - 4);
    for (int idx = tid; idx < 3 * GYh * (GXh / 4); idx += NTHREADS) {
      int p   = idx / (GYh * (GXh / 4));
      int rem = idx - p * (GYh * (GXh / 4));
      int r   = rem / (GXh / 4);
      int c4  = rem - r * (GXh / 4);
      async_copy_b128(base + (long long)p * PLANE + r * Wd + c4 * 4,
                      &S[p * NG + r * GXh + c4 * 4]);
    }
  } else {
    // generic b32 path with per-element reflect indexing
    for (int idx = tid; idx < 3 * NG; idx += NTHREADS) {
      int p   = idx / NG;
      int rem = idx - p * NG;
      int gy  = rem / GXh;
      int gx  = rem - gy * GXh;
      int rr  = refl(row0 - 4 + gy);
      int cc  = refl(col0 - 4 + gx);
      async_copy_b32(img + (long long)p * PLANE + rr * Wd + cc, &S[idx]);
    }
  }
  wait_async_zero();
  __syncthreads();

  // ---- phase 1b: gray (vectorized, in place over R plane) ----
  {
    v4f* S4 = (v4f*)S;
    for (int idx = tid; idx < NG / 4; idx += NTHREADS) {
      v4f r = S4[idx];
      v4f g = S4[(NG / 4) + idx];
      v4f b = S4[2 * (NG / 4) + idx];
      v4f y = 0.2989f * r + 0.587f * g + 0.114f * b;
      S4[idx] = y;  // each thread only touches its own chunk
    }
  }
  __syncthreads();

  // ---- phase 2: 3x3 Gaussian blur -> region1 ----
  for (int idx = tid; idx < NB; idx += NTHREADS) {
    int by = idx / BXh;
    int bx = idx - by * BXh;
    const float* g0 = &S[by * GXh + bx];
    float r0 = GW0 * g0[0]        + GW1 * g0[1]          + GW0 * g0[2];
    float r1 = GW0 * g0[GXh]      + GW1 * g0[GXh + 1]    + GW0 * g0[GXh + 2];
    float r2 = GW0 * g0[2 * GXh]  + GW1 * g0[2 * GXh + 1]+ GW0 * g0[2 * GXh + 2];
    S[NG + idx] = GW0 * r0 + GW1 * r1 + GW0 * r2;
  }
  __syncthreads();

  // ---- phase 3: Sobel + magnitude (-inf outside image) -> region2 ----
  for (int idx = tid; idx < NM; idx += NTHREADS) {
    int my = idx / MXh;
    int mx = idx - my * MXh;
    int rr = row0 - 2 + my;
    int cc = col0 - 2 + mx;
    float m;
    if (rr >= 0 && rr < Hd && cc >= 0 && cc < Wd) {
      const float* b = &S[NG + my * BXh + mx];
      float a0 = b[0],           a1 = b[1],           a2 = b[2];
      float d0 = b[BXh],                              d2 = b[BXh + 2];
      float e0 = b[2 * BXh],     e1 = b[2 * BXh + 1], e2 = b[2 * BXh + 2];
      float gx_ = (a0 - a2) + 2.0f * (d0 - d2) + (e0 - e2);
      float gy_ = (a0 + 2.0f * a1 + a2) - (e0 + 2.0f * e1 + e2);
      m = sqrtf(gx_ * gx_ + gy_ * gy_);
    } else {
      m = -__builtin_inff();  // maxpool pad value
    }
    S[2 * NG + idx] = m;
  }
  __syncthreads();

  // ---- phase 4a: horizontal 5-wide max -> region0 (gray dead) ----
  for (int idx = tid; idx < NHM; idx += NTHREADS) {
    int hy = idx / TX;
    int x  = idx - hy * TX;
    const float* mr = &S[2 * NG + hy * MXh + x];
    float m = mr[0];
    m = fmaxf(m, mr[1]); m = fmaxf(m, mr[2]);
    m = fmaxf(m, mr[3]); m = fmaxf(m, mr[4]);
    S[idx] = m;
  }
  __syncthreads();

  // ---- phase 4b: vertical 5-wide max (vectorized) -> non-temporal global store ----
  for (int idx = tid; idx < (TY * TX) / 4; idx += NTHREADS) {
    int y  = idx / (TX / 4);
    int x4 = (idx - y * (TX / 4)) * 4;
    const v4f* c = (const v4f*)&S[y * TX + x4];
    v4f m = c[0];
    m = max4(m, c[1 * (TX / 4)]);
    m = max4(m, c[2 * (TX / 4)]);
    m = max4(m, c[3 * (TX / 4)]);
    m = max4(m, c[4 * (TX / 4)]);
    nt_store4(out + (long long)(row0 + y) * Wd + (col0 + x4), m);
  }
}

// ---------------- host entry ----------------
extern "C" void kernel_launch(void* const* d_in, const int* in_sizes, int n_in,
                              void* d_out, int out_size, void* d_ws, size_t ws_size,
                              hipStream_t stream) {
  (void)in_sizes; (void)n_in; (void)out_size; (void)d_ws; (void)ws_size;
  const float* img = (const float*)d_in[0];   // [3,4096,4096] f32
  float* out = (float*)d_out;                 // [4096,4096] f32
  dim3 grid(Wd / TX, Hd / TY);                // 32 x 128 blocks
  sobel_fused<<<grid, NTHREADS, 0, stream>>>(img, out);
}